// PhongSurfacePy3d_12463995093941
// MI455X (gfx1250) — compile-verified
//
#include <hip/hip_runtime.h>

typedef __attribute__((ext_vector_type(2))) float v2f;
typedef __attribute__((ext_vector_type(8))) float v8f;

#define N_VERTS   5000
#define N_FACES   10000
#define N_PTS_TOT 16384            // BATCH * N_PTS
#define N_CHUNKS  (N_FACES / 16)   // 625

// ---------------------------------------------------------------------------
// Kernel 1: triangle centers + squared norms -> float4 {x,y,z,|c|^2}
// ---------------------------------------------------------------------------
__global__ void centers_kernel(const float* __restrict__ mesh_V,
                               const int*   __restrict__ mesh_F,
                               float4*      __restrict__ cent) {
  int f = blockIdx.x * blockDim.x + threadIdx.x;
  if (f >= N_FACES) return;
  int i0 = mesh_F[3 * f + 0], i1 = mesh_F[3 * f + 1], i2 = mesh_F[3 * f + 2];
  const float k = 1.0f / 3.0f;
  float x = (mesh_V[3 * i0 + 0] + mesh_V[3 * i1 + 0] + mesh_V[3 * i2 + 0]) * k;
  float y = (mesh_V[3 * i0 + 1] + mesh_V[3 * i1 + 1] + mesh_V[3 * i2 + 1]) * k;
  float z = (mesh_V[3 * i0 + 2] + mesh_V[3 * i1 + 2] + mesh_V[3 * i2 + 2]) * k;
  cent[f] = make_float4(x, y, z, x * x + y * y + z * z);
}

// ---------------------------------------------------------------------------
// Kernel 2: KNN (K=1) via V_WMMA_F32_16X16X4_F32 with fused argmin.
// Centers staged once per block into LDS (160000 B <= 320 KB WGP LDS), so
// each block reads the table from L2 exactly once instead of once per wave.
// One wave = 16 query points (A tile, loaded once); loop 625 chunks of 16
// centers (B tile from LDS, prefetched one chunk ahead so ds_load overlaps
// the WMMA + argmin of the previous chunk).
// D[m][n] = v_m . c_n ; d2 = |c|^2 - 2*dot (row term constant for argmin).
// Each lane owns column n = lane&15 and rows r(+8*half) -> per-lane running
// min, then shfl_xor min-reduce across the 16 lanes of each half.
// ---------------------------------------------------------------------------
__global__ void knn_wmma_kernel(const float*  __restrict__ verts,
                                const float4* __restrict__ cent,
                                int*          __restrict__ fid,
                                float*        __restrict__ out_fidx) {
  extern __shared__ float4 scent[];  // N_FACES entries (dynamic LDS)

  // Cooperative stage: coalesced global_load_b128 -> ds_store_b128.
  for (int i = threadIdx.x; i < N_FACES; i += blockDim.x) scent[i] = cent[i];
  __syncthreads();

  const int lane = threadIdx.x & 31;
  const int wave = threadIdx.x >> 5;
  const int tile = blockIdx.x * (blockDim.x >> 5) + wave;  // 0..1023
  const int half = lane >> 4;
  const int l    = lane & 15;

  // A tile: lane l holds row M=l; low half: (K0,K1)=(x,y), high half: (K2,K3)=(z,0)
  v2f a;
  {
    const float* q = verts + 3 * (tile * 16 + l);
    float qx = q[0], qy = q[1], qz = q[2];
    a.x = half ? qz : qx;
    a.y = half ? 0.0f : qy;
  }

  float minv[8];
  int   mini[8];
#pragma unroll
  for (int r = 0; r < 8; ++r) { minv[r] = 3.4e38f; mini[r] = 0; }

  float4 c = scent[l];  // chunk 0, this lane's column
  for (int ch = 0; ch < N_CHUNKS; ++ch) {
    int chn = ch + 1;
    if (chn == N_CHUNKS) chn = 0;               // uniform scalar select
    float4 cnext = scent[chn * 16 + l];         // prefetch next chunk (LDS)

    v2f b;
    b.x = half ? c.z : c.x;   // B tile: transpose-symmetric K striping
    b.y = half ? 0.0f : c.y;
    v8f acc = {};
    acc = __builtin_amdgcn_wmma_f32_16x16x4_f32(
        /*neg_a=*/false, a, /*neg_b=*/false, b,
        /*c_mod=*/(short)0, acc, /*reuse_a=*/false, /*reuse_b=*/false);

    const int   n  = ch * 16 + l;
    const float cn = c.w;
#pragma unroll
    for (int r = 0; r < 8; ++r) {
      float d2 = cn - 2.0f * acc[r];
      if (d2 < minv[r]) { minv[r] = d2; mini[r] = n; }
    }
    c = cnext;
  }

  // min-reduce across the 16 lanes of each half (xor masks 1,2,4,8 stay in half)
#pragma unroll
  for (int mask = 1; mask < 16; mask <<= 1) {
#pragma unroll
    for (int r = 0; r < 8; ++r) {
      float ov = __shfl_xor(minv[r], mask, 32);
      int   oi = __shfl_xor(mini[r], mask, 32);
      if (ov < minv[r] || (ov == minv[r] && oi < mini[r])) {
        minv[r] = ov; mini[r] = oi;
      }
    }
  }

  if (l == 0) {
#pragma unroll
    for (int r = 0; r < 8; ++r) {
      int row = r + half * 8;           // D layout: VGPR r = row r / r+8 by half
      int p   = tile * 16 + row;
      fid[p]      = mini[r];
      out_fidx[p] = (float)mini[r];
    }
  }
}

// ---------------------------------------------------------------------------
// Kernel 3: per-point correspondence refine: 4 x (50-step unrolled Adam) with
// analytic gradient of mean((S*(P(w) + d2*normalize(n(w))) - S*q)^2).
// ---------------------------------------------------------------------------
__global__ void refine_kernel(const float* __restrict__ verts,
                              const float* __restrict__ mesh_V,
                              const float* __restrict__ mesh_N,
                              const int*   __restrict__ mesh_F,
                              const int*   __restrict__ fid,
                              float*       __restrict__ out_vw,
                              float*       __restrict__ out_mask) {
  int p = blockIdx.x * blockDim.x + threadIdx.x;
  if (p >= N_PTS_TOT) return;

  const float S    = 10.0f;
  const float C2   = 2.0f / (float)(N_PTS_TOT * 3);  // d/dx of mean((x-t)^2)
  const float GS   = C2 * S * S;                     // full gradient scale
  const float LR   = 0.01f;
  const float cB1  = 0.9f, cB2 = 0.999f;
  const float EPSc = 1e-8f;

  float qx = verts[3 * p + 0], qy = verts[3 * p + 1], qz = verts[3 * p + 2];
  int f  = fid[p];
  int i0 = mesh_F[3 * f + 0], i1 = mesh_F[3 * f + 1], i2 = mesh_F[3 * f + 2];

  float V0x = mesh_V[3 * i0], V0y = mesh_V[3 * i0 + 1], V0z = mesh_V[3 * i0 + 2];
  float V1x = mesh_V[3 * i1], V1y = mesh_V[3 * i1 + 1], V1z = mesh_V[3 * i1 + 2];
  float V2x = mesh_V[3 * i2], V2y = mesh_V[3 * i2 + 1], V2z = mesh_V[3 * i2 + 2];
  float N0x = mesh_N[3 * i0], N0y = mesh_N[3 * i0 + 1], N0z = mesh_N[3 * i0 + 2];
  float N1x = mesh_N[3 * i1], N1y = mesh_N[3 * i1 + 1], N1z = mesh_N[3 * i1 + 2];
  float N2x = mesh_N[3 * i2], N2y = mesh_N[3 * i2 + 1], N2z = mesh_N[3 * i2 + 2];

  // dP/dw0 = V0-V2, dP/dw1 = V1-V2 ; dn/dw0 = N0-N2, dn/dw1 = N1-N2
  float A0x = V0x - V2x, A0y = V0y - V2y, A0z = V0z - V2z;
  float A1x = V1x - V2x, A1y = V1y - V2y, A1z = V1z - V2z;
  float E0x = N0x - N2x, E0y = N0y - N2y, E0z = N0z - N2z;
  float E1x = N1x - N2x, E1y = N1y - N2y, E1z = N1z - N2z;

  float vw0 = 1.0f / 3.0f, vw1 = 1.0f / 3.0f;

#pragma unroll 1
  for (int outer = 0; outer < 4; ++outer) {
    // d0 = || interp(tri_V, vw) - q ||
    float wc = 1.0f - vw0 - vw1;
    float Px = vw0 * V0x + vw1 * V1x + wc * V2x;
    float Py = vw0 * V0y + vw1 * V1y + wc * V2y;
    float Pz = vw0 * V0z + vw1 * V1z + wc * V2z;
    float dx = Px - qx, dy = Py - qy, dz = Pz - qz;
    float d0 = sqrtf(dx * dx + dy * dy + dz * dz);

    float dl0 = 0.0f, dl1 = 0.0f, dl2 = d0;
    float m0 = 0, m1 = 0, m2 = 0, v0 = 0, v1 = 0, v2 = 0;
    float b1p = 1.0f, b2p = 1.0f;

#pragma unroll 1
    for (int t = 0; t < 50; ++t) {
      b1p *= cB1; b2p *= cB2;
      float w0 = vw0 + dl0, w1 = vw1 + dl1, w2 = 1.0f - w0 - w1;

      float Qx = w0 * V0x + w1 * V1x + w2 * V2x;
      float Qy = w0 * V0y + w1 * V1y + w2 * V2y;
      float Qz = w0 * V0z + w1 * V1z + w2 * V2z;
      float nx = w0 * N0x + w1 * N1x + w2 * N2x;
      float ny = w0 * N0y + w1 * N1y + w2 * N2y;
      float nz = w0 * N0z + w1 * N1z + w2 * N2z;

      float nn  = sqrtf(nx * nx + ny * ny + nz * nz);
      nn        = fmaxf(nn, 1e-12f);
      float inn = 1.0f / nn;
      float ux = nx * inn, uy = ny * inn, uz = nz * inn;

      // residual / S
      float rx = Qx + dl2 * ux - qx;
      float ry = Qy + dl2 * uy - qy;
      float rz = Qz + dl2 * uz - qz;

      // g2 = GS * r . u
      float g2 = GS * (rx * ux + ry * uy + rz * uz);

      // du/dw0 . E0 = (E0 - u (u.E0)) / nn
      float ub0 = ux * E0x + uy * E0y + uz * E0z;
      float d0x = A0x + dl2 * (E0x - ux * ub0) * inn;
      float d0y = A0y + dl2 * (E0y - uy * ub0) * inn;
      float d0z = A0z + dl2 * (E0z - uz * ub0) * inn;
      float g0  = GS * (rx * d0x + ry * d0y + rz * d0z);

      float ub1 = ux * E1x + uy * E1y + uz * E1z;
      float d1x = A1x + dl2 * (E1x - ux * ub1) * inn;
      float d1y = A1y + dl2 * (E1y - uy * ub1) * inn;
      float d1z = A1z + dl2 * (E1z - uz * ub1) * inn;
      float g1  = GS * (rx * d1x + ry * d1y + rz * d1z);

      // Adam
      m0 = cB1 * m0 + (1.0f - cB1) * g0;  v0 = cB2 * v0 + (1.0f - cB2) * g0 * g0;
      m1 = cB1 * m1 + (1.0f - cB1) * g1;  v1 = cB2 * v1 + (1.0f - cB2) * g1 * g1;
      m2 = cB1 * m2 + (1.0f - cB1) * g2;  v2 = cB2 * v2 + (1.0f - cB2) * g2 * g2;
      float ib1 = 1.0f / (1.0f - b1p), ib2 = 1.0f / (1.0f - b2p);
      dl0 -= LR * (m0 * ib1) / (sqrtf(v0 * ib2) + EPSc);
      dl1 -= LR * (m1 * ib1) / (sqrtf(v1 * ib2) + EPSc);
      dl2 -= LR * (m2 * ib1) / (sqrtf(v2 * ib2) + EPSc);
    }
    vw0 += dl0;
    vw1 += dl1;
  }

  out_vw[2 * p + 0] = vw0;
  out_vw[2 * p + 1] = vw1;
  out_mask[p]       = 0.0f;
}

// ---------------------------------------------------------------------------
extern "C" void kernel_launch(void* const* d_in, const int* in_sizes, int n_in,
                              void* d_out, int out_size, void* d_ws, size_t ws_size,
                              hipStream_t stream) {
  const float* verts  = (const float*)d_in[0];  // [2,8192,3]
  const float* mesh_V = (const float*)d_in[1];  // [5000,3]
  const float* mesh_N = (const float*)d_in[2];  // [5000,3]
  const int*   mesh_F = (const int*)d_in[3];    // [10000,3]

  float* out = (float*)d_out;  // [fidx:16384][vw:32768][mask:16384]

  float4* cent = (float4*)d_ws;                               // 160000 B
  int*    fid  = (int*)((char*)d_ws + N_FACES * sizeof(float4));

  centers_kernel<<<(N_FACES + 255) / 256, 256, 0, stream>>>(mesh_V, mesh_F, cent);

  // 1024 tiles of 16 points; 8 waves/block -> 128 blocks, all waves full.
  // Dynamic LDS: full center table per block (160000 B <= 320 KB WGP LDS).
  knn_wmma_kernel<<<N_PTS_TOT / 16 / 8, 256, N_FACES * sizeof(float4), stream>>>(
      verts, cent, fid, out);

  refine_kernel<<<(N_PTS_TOT + 255) / 256, 256, 0, stream>>>(
      verts, mesh_V, mesh_N, mesh_F, fid,
      out + N_PTS_TOT, out + 3 * N_PTS_TOT);
}